// Shared_53377853554906
// MI455X (gfx1250) — compile-verified
//
#include <hip/hip_runtime.h>
#include <hip/hip_bf16.h>
#include <math.h>

#if defined(__HIP_DEVICE_COMPILE__)
#if !__has_builtin(__builtin_amdgcn_wmma_f32_16x16x4_f32)
#error "wmma f32 16x16x4 builtin not available on this device toolchain"
#endif
#endif

// ---------------- problem constants ----------------
#define BB   16          // batch
#define FF   36          // filters
#define CHN  22          // EEG channels
#define TIN  1001        // input time
#define RF_  65          // temporal receptive field
#define TT_V 937         // TIN - RF_ + 1 (valid temporal outputs)
#define TT_S 944         // padded stride (mult of 16)
#define KS_  15          // crop conv kernel
#define PU_V 309         // pooled length needed (u <= 308)
#define PU_S 320         // padded stride
#define EU_V 295         // E length needed (u <= 294)
#define EU_S 304         // padded stride (19 tiles of 16)
#define NQ   167         // crops per phase (q in [0,166])
#define EPSF 1e-5f

typedef float v2f __attribute__((ext_vector_type(2)));
typedef float v8f __attribute__((ext_vector_type(8)));

__device__ __forceinline__ v8f wmma4(v2f a, v2f b, v8f c) {
#if defined(__HIP_DEVICE_COMPILE__)
  // D = A(16x4,f32) * B(4x16,f32) + C(16x16,f32)
  return __builtin_amdgcn_wmma_f32_16x16x4_f32(false, a, false, b, (short)0, c,
                                               false, false);
#else
  (void)a; (void)b;
  return c;  // host pass only parses this
#endif
}

__device__ __forceinline__ int mini(int a, int b) { return a < b ? a : b; }

__device__ __forceinline__ float eluf(float x) {
  return x > 0.f ? x : (__expf(x) - 1.f);
}

__device__ __forceinline__ float bnf(float x, float g, float b, float m, float v) {
  return (x - m) * (g * (1.f / sqrtf(v + EPSF))) + b;
}

// ---------------------------------------------------------------------------
// K1: temporal conv + bias + BN + ELU.
//   A1[b][f][ch][t] = elu(bn_t(sum_k w_t[f,k] * x[b,ch,t+k] + b_t[f]))
// GEMM per (b,ch): M=36 (3 m-tiles), K=65, N=937 (59 tiles). 1 wave / C tile.
// Addresses are clamped (not predicated): garbage A-rows / B-cols only feed
// outputs discarded at the store; only the K tail (k=65..67) is zero-masked.
// ---------------------------------------------------------------------------
__global__ __launch_bounds__(32) void k1_temporal(
    const float* __restrict__ x, const float* __restrict__ w_t,
    const float* __restrict__ b_t, const float* __restrict__ g,
    const float* __restrict__ bt, const float* __restrict__ mn,
    const float* __restrict__ vr, float* __restrict__ A1) {
  const int lane = threadIdx.x & 31;
  const int l15  = lane & 15;
  const int hi   = lane >> 4;           // 0: lanes 0-15, 1: lanes 16-31
  const int nt   = blockIdx.x % 59;
  const int mt   = blockIdx.x / 59;     // 0..2
  const int ch   = blockIdx.y;
  const int b    = blockIdx.z;
  const int col  = nt * 16 + l15;       // output t
  const int arow = mt * 16 + l15;       // filter f for A fragment

  const int col_c  = mini(col, TT_V - 1);
  const int arow_c = mini(arow, FF - 1);
  const float* __restrict__ xrow = x + ((size_t)b * CHN + ch) * TIN + col_c;
  const float* __restrict__ wrow = w_t + (size_t)arow_c * RF_;

  v8f acc0 = {}, acc1 = {};
  // main K loop: k = 0..63, two interleaved accumulator chains
  for (int kk = 0; kk < 64; kk += 8) {
    v2f a0, b0, a1, b1;
    const int ka0 = kk + hi * 2;
    const int kb0 = kk + hi;
    a0.x = wrow[ka0];         a0.y = wrow[ka0 + 1];
    b0.x = xrow[kb0];         b0.y = xrow[kb0 + 2];
    a1.x = wrow[ka0 + 4];     a1.y = wrow[ka0 + 5];
    b1.x = xrow[kb0 + 4];     b1.y = xrow[kb0 + 6];
    acc0 = wmma4(a0, b0, acc0);
    acc1 = wmma4(a1, b1, acc1);
  }
  // K tail: k = 64 valid, 65..67 zero (mask via multiply, loads stay plain)
  {
    const float m0 = (hi == 0) ? 1.f : 0.f;
    v2f a, bf;
    a.x  = wrow[64] * m0;  a.y  = 0.f;
    bf.x = xrow[64] * m0;  bf.y = 0.f;
    acc0 = wmma4(a, bf, acc0);
  }

  const bool colv = (col < TT_V);
#pragma unroll
  for (int j = 0; j < 8; ++j) {
    const int f = mt * 16 + j + hi * 8;   // C/D layout: M = j (+8 for hi half)
    if (f < FF && colv) {
      float val = acc0[j] + acc1[j] + b_t[f];
      val = eluf(bnf(val, g[f], bt[f], mn[f], vr[f]));
      A1[(((size_t)b * FF + f) * CHN + ch) * TT_S + col] = val;
    }
  }
}

// ---------------------------------------------------------------------------
// K2: spatial conv + bias + BN + ELU.
//   H[b][f2][t] = elu(bn_s(sum_{f1,ch} w_s[f2,f1,ch] * A1[b,f1,ch,t] + b_s))
// GEMM per b: M=36, K=792 (row index (f1*22+ch) is contiguous in A1), N=937.
// K divisible by 8 -> branch-free, mask-free inner loop.
// ---------------------------------------------------------------------------
__global__ __launch_bounds__(32) void k2_spatial(
    const float* __restrict__ A1, const float* __restrict__ w_s,
    const float* __restrict__ b_s, const float* __restrict__ g,
    const float* __restrict__ bt, const float* __restrict__ mn,
    const float* __restrict__ vr, float* __restrict__ H) {
  const int lane = threadIdx.x & 31;
  const int l15  = lane & 15;
  const int hi   = lane >> 4;
  const int nt   = blockIdx.x % 59;
  const int mt   = blockIdx.x / 59;
  const int b    = blockIdx.y;
  const int col  = nt * 16 + l15;
  const int arow = mt * 16 + l15;

  const int col_c  = mini(col, TT_V - 1);
  const int arow_c = mini(arow, FF - 1);
  const float* __restrict__ Bb   = A1 + (size_t)b * (FF * CHN) * TT_S + col_c;
  const float* __restrict__ wrow = w_s + (size_t)arow_c * (FF * CHN);

  v8f acc0 = {}, acc1 = {};
  for (int kk = 0; kk < FF * CHN; kk += 8) {   // 792 = 8*99
    v2f a0, b0, a1, b1;
    const int ka = kk + hi * 2;
    const int kb = kk + hi;
    a0.x = wrow[ka];      a0.y = wrow[ka + 1];
    b0.x = Bb[(size_t)kb * TT_S];
    b0.y = Bb[(size_t)(kb + 2) * TT_S];
    a1.x = wrow[ka + 4];  a1.y = wrow[ka + 5];
    b1.x = Bb[(size_t)(kb + 4) * TT_S];
    b1.y = Bb[(size_t)(kb + 6) * TT_S];
    acc0 = wmma4(a0, b0, acc0);
    acc1 = wmma4(a1, b1, acc1);
  }

  const bool colv = (col < TT_V);
#pragma unroll
  for (int j = 0; j < 8; ++j) {
    const int f = mt * 16 + j + hi * 8;
    if (f < FF && colv) {
      float val = acc0[j] + acc1[j] + b_s[f];
      val = eluf(bnf(val, g[f], bt[f], mn[f], vr[f]));
      H[((size_t)b * FF + f) * TT_S + col] = val;
    }
  }
}

// ---------------------------------------------------------------------------
// K3: three-phase avg-pool3.  P[r][b][f][u] = mean(H[3u+r .. 3u+r+2])
// ---------------------------------------------------------------------------
__global__ void k3_pool(const float* __restrict__ H, float* __restrict__ P) {
  const int total = 3 * BB * FF * PU_V;
  int i = blockIdx.x * blockDim.x + threadIdx.x;
  if (i >= total) return;
  const int u = i % PU_V;
  int t = i / PU_V;
  const int f = t % FF; t /= FF;
  const int b = t % BB;
  const int r = t / BB;
  const float* h = H + ((size_t)b * FF + f) * TT_S + 3 * u + r;
  P[(((size_t)r * BB + b) * FF + f) * PU_S + u] =
      (h[0] + h[1] + h[2]) * (1.f / 3.f);
}

// ---------------------------------------------------------------------------
// K4: phase conv1d + bias + BN_c + ELU.
//   E[r][b][f2][u] = elu(bn_c(sum_{f1,k} w_c[f2,f1,k] * P[r,b,f1,u+k] + b_c))
// GEMM per (r,b): M=36, K=540, N=295 (19 tiles); B reads shifted P windows.
// Clamped col: u+k <= 308 < PU_S stays in the padded (written) P row.
// ---------------------------------------------------------------------------
__global__ __launch_bounds__(32) void k4_convc(
    const float* __restrict__ P, const float* __restrict__ w_c,
    const float* __restrict__ b_c, const float* __restrict__ g,
    const float* __restrict__ bt, const float* __restrict__ mn,
    const float* __restrict__ vr, float* __restrict__ E) {
  const int lane = threadIdx.x & 31;
  const int l15  = lane & 15;
  const int hi   = lane >> 4;
  const int nt   = blockIdx.x % 19;
  const int mt   = blockIdx.x / 19;
  const int b    = blockIdx.y;
  const int r    = blockIdx.z;
  const int col  = nt * 16 + l15;     // output u
  const int arow = mt * 16 + l15;

  const int col_c  = mini(col, EU_V - 1);
  const int arow_c = mini(arow, FF - 1);
  const float* __restrict__ Pb   = P + ((size_t)r * BB + b) * FF * PU_S + col_c;
  const float* __restrict__ wrow = w_c + (size_t)arow_c * (FF * KS_);

  v8f acc0 = {}, acc1 = {};
  // 540 = 8*67 + 4: paired main loop + one single step
  for (int kk = 0; kk < 536; kk += 8) {
    v2f a0, b0, a1, b1;
    const int ka  = kk + hi * 2;
    const int kb0 = kk + hi;
    a0.x = wrow[ka];      a0.y = wrow[ka + 1];
    a1.x = wrow[ka + 4];  a1.y = wrow[ka + 5];
    {
      const int k0 = kb0,     f1a = k0 / KS_, kma = k0 % KS_;
      const int k1 = kb0 + 2, f1b = k1 / KS_, kmb = k1 % KS_;
      b0.x = Pb[(size_t)f1a * PU_S + kma];
      b0.y = Pb[(size_t)f1b * PU_S + kmb];
    }
    {
      const int k0 = kb0 + 4, f1a = k0 / KS_, kma = k0 % KS_;
      const int k1 = kb0 + 6, f1b = k1 / KS_, kmb = k1 % KS_;
      b1.x = Pb[(size_t)f1a * PU_S + kma];
      b1.y = Pb[(size_t)f1b * PU_S + kmb];
    }
    acc0 = wmma4(a0, b0, acc0);
    acc1 = wmma4(a1, b1, acc1);
  }
  {
    const int kk = 536;
    v2f a, bf;
    const int ka  = kk + hi * 2;
    const int kb0 = kk + hi;
    a.x = wrow[ka];  a.y = wrow[ka + 1];
    const int k0 = kb0,     f1a = k0 / KS_, kma = k0 % KS_;
    const int k1 = kb0 + 2, f1b = k1 / KS_, kmb = k1 % KS_;
    bf.x = Pb[(size_t)f1a * PU_S + kma];
    bf.y = Pb[(size_t)f1b * PU_S + kmb];
    acc0 = wmma4(a, bf, acc0);
  }

  const bool colv = (col < EU_V);
#pragma unroll
  for (int j = 0; j < 8; ++j) {
    const int f = mt * 16 + j + hi * 8;
    if (f < FF && colv) {
      float val = acc0[j] + acc1[j] + b_c[f];
      val = eluf(bnf(val, g[f], bt[f], mn[f], vr[f]));
      E[(((size_t)r * BB + b) * FF + f) * EU_S + col] = val;
    }
  }
}

// ---------------------------------------------------------------------------
// K5: fold (pool2 + FC + crop-sum) into G[f][u][o] (phase-independent):
//   G[f,u,o] = (1/3) * sum_{j=0..2} sum_{v=0..42} [0 <= u-3v-j <= 166]
//                  * w_fc[o, f*43+v]
// ---------------------------------------------------------------------------
__global__ void k5_buildG(const float* __restrict__ w_fc, float* __restrict__ G) {
  const int total = FF * EU_V * 4;
  int i = blockIdx.x * blockDim.x + threadIdx.x;
  if (i >= total) return;
  const int o = i % 4;
  int t = i / 4;
  const int u = t % EU_V;
  const int f = t / EU_V;
  float s = 0.f;
  for (int v = 0; v < 43; ++v) {
    const float w = w_fc[o * (FF * 43) + f * 43 + v];
#pragma unroll
    for (int j = 0; j < 3; ++j) {
      const int q = u - 3 * v - j;
      if (q >= 0 && q <= (NQ - 1)) s += w;
    }
  }
  G[((size_t)f * EU_S + u) * 4 + o] = s * (1.f / 3.f);
}

// ---------------------------------------------------------------------------
// K6: final contraction + average over 501 crops.
//   out[b,o] = b_fc[o] + (1/501) * sum_{r,f,u} E[r,b,f,u] * G[f,u,o]
// ---------------------------------------------------------------------------
__global__ __launch_bounds__(256) void k6_final(
    const float* __restrict__ E, const float* __restrict__ G,
    const float* __restrict__ b_fc, float* __restrict__ out) {
  const int b = blockIdx.x;
  const int o = blockIdx.y;
  const int tid = threadIdx.x;
  __shared__ float red[256];

  const int span = 3 * FF * EU_V;   // 31860
  float s = 0.f;
  for (int i = tid; i < span; i += 256) {
    const int u = i % EU_V;
    int t = i / EU_V;
    const int f = t % FF;
    const int r = t / FF;
    s += E[(((size_t)r * BB + b) * FF + f) * EU_S + u] *
         G[((size_t)f * EU_S + u) * 4 + o];
  }
  red[tid] = s;
  __syncthreads();
  for (int w = 128; w > 0; w >>= 1) {
    if (tid < w) red[tid] += red[tid + w];
    __syncthreads();
  }
  if (tid == 0) out[b * 4 + o] = b_fc[o] + red[0] * (1.f / 501.f);
}

// ---------------------------------------------------------------------------
extern "C" void kernel_launch(void* const* d_in, const int* in_sizes, int n_in,
                              void* d_out, int out_size, void* d_ws,
                              size_t ws_size, hipStream_t stream) {
  const float* x      = (const float*)d_in[0];
  const float* w_t    = (const float*)d_in[1];
  const float* b_t    = (const float*)d_in[2];
  const float* bn_t_g = (const float*)d_in[3];
  const float* bn_t_b = (const float*)d_in[4];
  const float* bn_t_m = (const float*)d_in[5];
  const float* bn_t_v = (const float*)d_in[6];
  const float* w_s    = (const float*)d_in[7];
  const float* b_s    = (const float*)d_in[8];
  const float* bn_s_g = (const float*)d_in[9];
  const float* bn_s_b = (const float*)d_in[10];
  const float* bn_s_m = (const float*)d_in[11];
  const float* bn_s_v = (const float*)d_in[12];
  const float* w_c    = (const float*)d_in[13];
  const float* b_c    = (const float*)d_in[14];
  const float* bn_c_g = (const float*)d_in[15];
  const float* bn_c_b = (const float*)d_in[16];
  const float* bn_c_m = (const float*)d_in[17];
  const float* bn_c_v = (const float*)d_in[18];
  const float* w_fc   = (const float*)d_in[19];
  const float* b_fc   = (const float*)d_in[20];
  float* out = (float*)d_out;

  // Workspace carve-up (floats). Total ~54.5 MB — L2-resident (192 MB).
  float* ws = (float*)d_ws;
  float* A1 = ws;                                   // 16*36*22*944
  float* H  = A1 + (size_t)BB * FF * CHN * TT_S;    // 16*36*944
  float* P  = H  + (size_t)BB * FF * TT_S;          // 3*16*36*320
  float* E  = P  + (size_t)3 * BB * FF * PU_S;      // 3*16*36*304
  float* G  = E  + (size_t)3 * BB * FF * EU_S;      // 36*304*4

  k1_temporal<<<dim3(59 * 3, CHN, BB), 32, 0, stream>>>(
      x, w_t, b_t, bn_t_g, bn_t_b, bn_t_m, bn_t_v, A1);

  k2_spatial<<<dim3(59 * 3, BB, 1), 32, 0, stream>>>(
      A1, w_s, b_s, bn_s_g, bn_s_b, bn_s_m, bn_s_v, H);

  {
    const int total = 3 * BB * FF * PU_V;
    k3_pool<<<(total + 255) / 256, 256, 0, stream>>>(H, P);
  }

  k4_convc<<<dim3(19 * 3, BB, 3), 32, 0, stream>>>(
      P, w_c, b_c, bn_c_g, bn_c_b, bn_c_m, bn_c_v, E);

  {
    const int total = FF * EU_V * 4;
    k5_buildG<<<(total + 255) / 256, 256, 0, stream>>>(w_fc, G);
  }

  k6_final<<<dim3(BB, 4, 1), 256, 0, stream>>>(E, G, b_fc, out);
}